// RGCNLayer_44478681318050
// MI455X (gfx1250) — compile-verified
//
#include <hip/hip_runtime.h>
#include <hip/hip_bf16.h>

typedef float v2f __attribute__((ext_vector_type(2)));
typedef float v8f __attribute__((ext_vector_type(8)));

#define IN_FEAT  256
#define OUT_FEAT 256

// ---------------------------------------------------------------------------
// Kernel 1: T = h @ W[r]   ([n_nodes,256] x [256,256] -> [n_nodes,256], f32)
// One wave (32 threads) per 16-row x 64-col output strip.
// Uses V_WMMA_F32_16X16X4_F32: A is 16x4 f32 (2 VGPRs), B is 4x16 f32
// (2 VGPRs), C/D is 16x16 f32 (8 VGPRs).  K loop: 256/4 = 64 wmma issues
// per 16x16 tile, A fragment reused across the 4 column tiles.
// ---------------------------------------------------------------------------
__global__ void __launch_bounds__(32)
rgcn_gemm_f32_wmma(const float* __restrict__ h,
                   const float* __restrict__ Wr,   // already offset to relation r
                   float* __restrict__ T,
                   int n_nodes) {
    const int lane    = threadIdx.x;          // 0..31 (wave32)
    const int rowbase = blockIdx.x * 16;      // node tile
    const int obase   = blockIdx.y * 64;      // 4 x 16 output columns

    const int half  = lane >> 4;              // 0 for lanes 0-15, 1 for 16-31
    const int l15   = lane & 15;
    const int khalf = half * 2;               // K offset within 4-wide slab

    // clamp row for safety (n_nodes is a multiple of 16 in this problem)
    int row = rowbase + l15;
    if (row >= n_nodes) row = n_nodes - 1;
    const float* hrow = h + (size_t)row * IN_FEAT;

    v8f acc0 = {}, acc1 = {}, acc2 = {}, acc3 = {};

    const int col0 = obase + l15;

    for (int k0 = 0; k0 < IN_FEAT; k0 += 4) {
        // A fragment: lane<16 holds (K=k0+0, k0+1), lane>=16 holds (k0+2, k0+3)
        v2f a = *(const v2f*)(hrow + k0 + khalf);

        const float* wk0 = Wr + (size_t)(k0 + khalf) * OUT_FEAT;
        const float* wk1 = wk0 + OUT_FEAT;

        // B fragments for the 4 column tiles (mirror of A layout, N=lane&15)
        v2f b0 = { wk0[col0 +  0], wk1[col0 +  0] };
        v2f b1 = { wk0[col0 + 16], wk1[col0 + 16] };
        v2f b2 = { wk0[col0 + 32], wk1[col0 + 32] };
        v2f b3 = { wk0[col0 + 48], wk1[col0 + 48] };

        acc0 = __builtin_amdgcn_wmma_f32_16x16x4_f32(false, a, false, b0,
                                                     (short)0, acc0, false, false);
        acc1 = __builtin_amdgcn_wmma_f32_16x16x4_f32(false, a, false, b1,
                                                     (short)0, acc1, false, false);
        acc2 = __builtin_amdgcn_wmma_f32_16x16x4_f32(false, a, false, b2,
                                                     (short)0, acc2, false, false);
        acc3 = __builtin_amdgcn_wmma_f32_16x16x4_f32(false, a, false, b3,
                                                     (short)0, acc3, false, false);
    }

    // C/D layout: VGPR j holds M=j (lanes 0-15) and M=j+8 (lanes 16-31), N=lane&15
    const int moff = half * 8;
#pragma unroll
    for (int j = 0; j < 8; ++j) {
        int r = rowbase + j + moff;
        if (r < n_nodes) {
            float* trow = T + (size_t)r * OUT_FEAT;
            trow[col0 +  0] = acc0[j];
            trow[col0 + 16] = acc1[j];
            trow[col0 + 32] = acc2[j];
            trow[col0 + 48] = acc3[j];
        }
    }
}

// ---------------------------------------------------------------------------
// Kernel 2: for edges with rel==rel_id:  out[dst] += T[src] * norm
// One 256-thread block sweeps a chunk of edges; one feature per lane,
// fully coalesced gather + global_atomic_add_f32 scatter (L2-resident).
// ---------------------------------------------------------------------------
__global__ void __launch_bounds__(OUT_FEAT)
rgcn_scatter(const float* __restrict__ T,
             const float* __restrict__ norm,
             const int*   __restrict__ src,
             const int*   __restrict__ dst,
             const int*   __restrict__ rel,
             float*       __restrict__ out,
             int rel_id, int n_edges, int edges_per_block) {
    const int tid = threadIdx.x;              // feature index 0..255
    const int e0  = blockIdx.x * edges_per_block;

    for (int i = 0; i < edges_per_block; ++i) {
        const int e = e0 + i;
        if (e >= n_edges) break;
        // prefetch the edge-stream ahead (global_prefetch_b8)
        if (e + 16 < n_edges) __builtin_prefetch(&rel[e + 16], 0, 0);
        if (rel[e] != rel_id) continue;       // uniform across the block
        const int   s  = src[e];
        const int   d  = dst[e];
        const float nv = norm[e];
        const float v  = T[(size_t)s * OUT_FEAT + tid] * nv;
        atomicAdd(&out[(size_t)d * OUT_FEAT + tid], v);
    }
}

// ---------------------------------------------------------------------------
// Fallback (only if workspace is too small): direct per-edge GEMV.
// ---------------------------------------------------------------------------
__global__ void __launch_bounds__(OUT_FEAT)
rgcn_naive(const float* __restrict__ h,
           const float* __restrict__ W,
           const float* __restrict__ norm,
           const int*   __restrict__ src,
           const int*   __restrict__ dst,
           const int*   __restrict__ rel,
           float*       __restrict__ out,
           int n_edges) {
    const int e = blockIdx.x;
    if (e >= n_edges) return;
    const int o = threadIdx.x;
    const int s = src[e], d = dst[e], r = rel[e];
    const float* hrow = h + (size_t)s * IN_FEAT;
    const float* wr   = W + (size_t)r * IN_FEAT * OUT_FEAT + o;
    float acc = 0.f;
#pragma unroll 8
    for (int k = 0; k < IN_FEAT; ++k)
        acc += hrow[k] * wr[(size_t)k * OUT_FEAT];
    atomicAdd(&out[(size_t)d * OUT_FEAT + o], acc * norm[e]);
}

// ---------------------------------------------------------------------------
// Host launcher
// ---------------------------------------------------------------------------
extern "C" void kernel_launch(void* const* d_in, const int* in_sizes, int n_in,
                              void* d_out, int out_size, void* d_ws, size_t ws_size,
                              hipStream_t stream) {
    const float* h    = (const float*)d_in[0];
    const float* W    = (const float*)d_in[1];
    const float* norm = (const float*)d_in[2];
    const int*   src  = (const int*)d_in[3];
    const int*   dst  = (const int*)d_in[4];
    const int*   rel  = (const int*)d_in[5];
    float*       out  = (float*)d_out;

    const int n_nodes  = in_sizes[0] / IN_FEAT;
    const int num_rels = in_sizes[1] / (IN_FEAT * OUT_FEAT);
    const int n_edges  = in_sizes[3];

    // out accumulates via atomics -> must start at zero every call
    hipMemsetAsync(d_out, 0, (size_t)out_size * sizeof(float), stream);

    const size_t t_bytes = (size_t)n_nodes * OUT_FEAT * sizeof(float);

    if (ws_size >= t_bytes) {
        float* T = (float*)d_ws;
        const int row_tiles = (n_nodes + 15) / 16;      // 1250
        const dim3 ggrid(row_tiles, OUT_FEAT / 64);     // (1250, 4)
        const int  epb = 128;
        const int  sgrid = (n_edges + epb - 1) / epb;   // 2500

        for (int r = 0; r < num_rels; ++r) {
            rgcn_gemm_f32_wmma<<<ggrid, 32, 0, stream>>>(
                h, W + (size_t)r * IN_FEAT * OUT_FEAT, T, n_nodes);
            rgcn_scatter<<<sgrid, OUT_FEAT, 0, stream>>>(
                T, norm, src, dst, rel, out, r, n_edges, epb);
        }
    } else {
        rgcn_naive<<<n_edges, OUT_FEAT, 0, stream>>>(
            h, W, norm, src, dst, rel, out, n_edges);
    }
}